// Wav2vecSemantic_38611755991827
// MI455X (gfx1250) — compile-verified
//
#include <hip/hip_runtime.h>
#include <hip/hip_bf16.h>

typedef __attribute__((ext_vector_type(2))) float v2f;
typedef __attribute__((ext_vector_type(8))) float v8f;
typedef __attribute__((ext_vector_type(4))) int   i4;

// Address-space-qualified pointee typedefs for the async-copy builtins.
typedef __attribute__((address_space(1))) i4  gas_i4;
typedef __attribute__((address_space(3))) i4  las_i4;
typedef __attribute__((address_space(1))) int gas_i32;
typedef __attribute__((address_space(3))) int las_i32;

static __host__ inline int cdiv_h(int a, int b) { return (a + b - 1) / b; }

// ---------------------------------------------------------------------------
// CDNA5 async copy helpers: GLOBAL_LOAD_ASYNC_TO_LDS_* (ASYNCcnt-tracked)
// ---------------------------------------------------------------------------
__device__ __forceinline__ void async_wait0() {
#if __has_builtin(__builtin_amdgcn_s_wait_asynccnt)
  __builtin_amdgcn_s_wait_asynccnt(0);
#else
  asm volatile("s_wait_asynccnt 0" ::: "memory");
#endif
}

__device__ __forceinline__ void async_ld_b128(float* lds, const float* g) {
  __builtin_amdgcn_global_load_async_to_lds_b128((gas_i4*)g, (las_i4*)lds, 0, 0);
}

__device__ __forceinline__ void async_ld_b32(float* lds, const float* g) {
  __builtin_amdgcn_global_load_async_to_lds_b32((gas_i32*)g, (las_i32*)lds, 0, 0);
}

// ---------------------------------------------------------------------------
// LDS-staged, double-buffered implicit-GEMM conv1d (Cin=512 layers).
// Block = 256 threads = 8 waves; block tile 64 (Cout) x 32 (cols = B*Tout);
// K staged in chunks of 32 through ping-pong LDS (padded stride 36 ->
// conflict-free, 16B-aligned rows for b128 async copies).
// One barrier per chunk; chunk i+1 DMA overlaps chunk i WMMA.
// Out-of-range *columns* are loaded from clamped addresses (garbage only
// reaches output columns that are never stored); only PAD-generated elements
// need real zero-fill.
// ---------------------------------------------------------------------------
template<int CIN, int KW, int PAD>
__global__ __launch_bounds__(256)
void conv_lds_wmma(const float* __restrict__ in, const float* __restrict__ w,
                   const float* __restrict__ bias, float* __restrict__ out,
                   int B, int Lin, int Cout, int stride, int Tout)
{
  constexpr int KDIM = CIN * KW;
  static_assert(KDIM % 32 == 0, "K chunking requires KDIM % 32 == 0");
  __shared__ float As[2][64][36];   // [buf][co_local][k_local]
  __shared__ float Bt[2][32][36];   // [buf][col_local][k_local]

  const int N = B * Tout;
  const int colBlocks = (N + 31) >> 5;
  const int rb = blockIdx.x / colBlocks;
  const int cb = blockIdx.x % colBlocks;
  const int coBase  = rb << 6;
  const int colBase = cb << 5;
  const int tid = threadIdx.x;

  // A staging: thread -> (row am, 8-col group aq); two b128 asyncs per chunk
  const int am = tid >> 2;
  const int aq = (tid & 3) << 3;
  const float* aptr = w + (size_t)(coBase + am) * KDIM + aq;

  // B staging: thread -> (col bcol, 4 k-elements at bk0); clamped if col OOR
  const int bcol = tid >> 3;
  const int bk0  = (tid & 7) << 2;
  const int gcol = colBase + bcol;
  const bool gcolOK = gcol < N;
  const int gb = gcolOK ? (gcol / Tout) : 0;
  const int gt = gcolOK ? (gcol % Tout) : 0;
  const int pbase = gt * stride - PAD;
  const float* inb = in + (size_t)gb * CIN * Lin;

  // compute mapping: wave wv -> 16x16 tile at (rowOff, colOff)
  const int wv   = tid >> 5;
  const int lane = tid & 31;
  const int half = lane >> 4;
  const int l16  = lane & 15;
  const int rowOff = (wv >> 1) << 4;   // 0,16,32,48
  const int colOff = (wv & 1) << 4;    // 0,16

  auto stage = [&](int kk, int buf) {
    async_ld_b128(&As[buf][am][aq],     aptr + kk);
    async_ld_b128(&As[buf][am][aq + 4], aptr + kk + 4);
    if (PAD == 0 || (pbase >= 0 && pbase + KW <= Lin)) {
      // Fast path: all 4 elements in range -> unguarded asyncs, no EXEC churn
#pragma unroll
      for (int q = 0; q < 4; ++q) {
        const int r  = kk + bk0 + q;
        const int ci = r / KW;
        const int kq = r - ci * KW;
        async_ld_b32(&Bt[buf][bcol][bk0 + q], inb + (size_t)ci * Lin + pbase + kq);
      }
    } else {
      // Boundary columns of padded conv: per-element guard + zero fill
#pragma unroll
      for (int q = 0; q < 4; ++q) {
        const int r  = kk + bk0 + q;
        const int ci = r / KW;
        const int kq = r - ci * KW;
        const int p  = pbase + kq;
        if ((unsigned)p < (unsigned)Lin)
          async_ld_b32(&Bt[buf][bcol][bk0 + q], inb + (size_t)ci * Lin + p);
        else
          Bt[buf][bcol][bk0 + q] = 0.f;
      }
    }
  };

  v8f acc = {};
  stage(0, 0);
  int buf = 0;
  for (int kk = 0; kk < KDIM; kk += 32) {
    async_wait0();        // chunk kk landed in LDS (this wave's asyncs)
    __syncthreads();      // all waves done staging kk and reading buf^1
    if (kk + 32 < KDIM) stage(kk + 32, buf ^ 1);   // DMA overlaps WMMA below
#pragma unroll
    for (int kc = 0; kc < 32; kc += 4) {
      v2f av = { As[buf][rowOff + l16][kc + half * 2],
                 As[buf][rowOff + l16][kc + half * 2 + 1] };
      v2f bv = { Bt[buf][colOff + l16][kc + half * 2],
                 Bt[buf][colOff + l16][kc + half * 2 + 1] };
      acc = __builtin_amdgcn_wmma_f32_16x16x4_f32(false, av, false, bv,
                                                  (short)0, acc, false, false);
    }
    buf ^= 1;
  }

  const int scol = colBase + colOff + l16;
  if (scol < N) {
    const int ob = scol / Tout, ot = scol % Tout;
#pragma unroll
    for (int v = 0; v < 8; ++v) {
      const int oc = coBase + rowOff + half * 8 + v;
      out[((size_t)ob * Cout + oc) * Tout + ot] = acc[v] + bias[oc];
    }
  }
}

// ---------------------------------------------------------------------------
// LDS-staged, double-buffered GEMM for ConvTranspose einsum 'bit,iok->botk'.
// rows r = o*KK+k (M = 6144), reduce over i (512), cols = (b,t) (2384).
// A[r][i] = ctw[i*M + r] (K-major in memory -> per-element b32 asyncs).
// Entirely guard-free staging (column clamping; no padding exists here).
// ---------------------------------------------------------------------------
__global__ __launch_bounds__(256)
void convT_lds_wmma(const float* __restrict__ cin, const float* __restrict__ ctw,
                    const float* __restrict__ ctb, float* __restrict__ hk,
                    int B, int C, int T, int KK)
{
  __shared__ float As[2][64][36];   // [buf][r_local][i_local]
  __shared__ float Bt[2][32][36];   // [buf][col_local][i_local]

  const int M = C * KK;
  const int N = B * T;
  const int colBlocks = (N + 31) >> 5;
  const int rb = blockIdx.x / colBlocks;
  const int cb = blockIdx.x % colBlocks;
  const int rBase   = rb << 6;
  const int colBase = cb << 5;
  const int tid = threadIdx.x;

  // A staging: thread -> (i_local ai, 8 consecutive rows at ar0)
  const int ai  = tid >> 3;        // 0..31
  const int ar0 = (tid & 7) << 3;  // 0..56

  // B staging (clamped columns)
  const int bcol = tid >> 3;
  const int bk0  = (tid & 7) << 2;
  const int gcol = colBase + bcol;
  const bool gcolOK = gcol < N;
  const int gb = gcolOK ? (gcol / T) : 0;
  const int gt = gcolOK ? (gcol % T) : 0;

  const int wv   = tid >> 5;
  const int lane = tid & 31;
  const int half = lane >> 4;
  const int l16  = lane & 15;
  const int rowOff = (wv >> 1) << 4;
  const int colOff = (wv & 1) << 4;

  auto stage = [&](int kk, int buf) {
    const float* acol = ctw + (size_t)(kk + ai) * M + rBase + ar0;
#pragma unroll
    for (int j = 0; j < 8; ++j)
      async_ld_b32(&As[buf][ar0 + j][ai], acol + j);
#pragma unroll
    for (int q = 0; q < 4; ++q)
      async_ld_b32(&Bt[buf][bcol][bk0 + q],
                   cin + ((size_t)gb * C + kk + bk0 + q) * T + gt);
  };

  v8f acc = {};
  stage(0, 0);
  int buf = 0;
  for (int kk = 0; kk < C; kk += 32) {
    async_wait0();
    __syncthreads();
    if (kk + 32 < C) stage(kk + 32, buf ^ 1);
#pragma unroll
    for (int kc = 0; kc < 32; kc += 4) {
      v2f av = { As[buf][rowOff + l16][kc + half * 2],
                 As[buf][rowOff + l16][kc + half * 2 + 1] };
      v2f bv = { Bt[buf][colOff + l16][kc + half * 2],
                 Bt[buf][colOff + l16][kc + half * 2 + 1] };
      acc = __builtin_amdgcn_wmma_f32_16x16x4_f32(false, av, false, bv,
                                                  (short)0, acc, false, false);
    }
    buf ^= 1;
  }

  const int scol = colBase + colOff + l16;
  if (scol < N) {
    const int ob = scol / T, ot = scol % T;
#pragma unroll
    for (int v = 0; v < 8; ++v) {
      const int r = rBase + rowOff + half * 8 + v;
      const int o = r / KK, k = r % KK;
      hk[(((size_t)ob * C + o) * T + ot) * KK + k] = acc[v] + ctb[o];
    }
  }
}

// ---------------------------------------------------------------------------
// Direct WMMA conv for layer 1 (Cin=1, K=10): tiny FLOPs, no staging needed.
// ---------------------------------------------------------------------------
__global__ __launch_bounds__(256)
void conv1_wmma(const float* __restrict__ in, const float* __restrict__ w,
                const float* __restrict__ bias, float* __restrict__ out,
                int B, int Lin, int Cout, int stride, int Tout)
{
  constexpr int KDIM = 10;
  const int N   = B * Tout;
  const int ntx = (N + 15) >> 4;
  const int nty = Cout >> 4;
  const int wave = blockIdx.x * (blockDim.x >> 5) + (threadIdx.x >> 5);
  if (wave >= nty * ntx) return;
  const int ty = wave / ntx, tx = wave % ntx;
  const int lane = threadIdx.x & 31;
  const int half = lane >> 4, l16 = lane & 15;
  const int khalf = half * 2;
  const int co  = (ty << 4) + l16;
  const int col = (tx << 4) + l16;
  const bool colOK = col < N;
  const int b = colOK ? (col / Tout) : 0;
  const int t = colOK ? (col % Tout) : 0;
  const float* wrow = w + (size_t)co * KDIM;
  const float* inb  = in + (size_t)b * Lin;

  v8f acc = {};
  for (int kk = 0; kk < KDIM; kk += 4) {
    const int r0 = kk + khalf, r1 = r0 + 1;
    float a0 = (r0 < KDIM) ? wrow[r0] : 0.f;
    float a1 = (r1 < KDIM) ? wrow[r1] : 0.f;
    float b0 = 0.f, b1 = 0.f;
    if (r0 < KDIM) { int p = t * stride + r0; if (p < Lin) b0 = inb[p]; }
    if (r1 < KDIM) { int p = t * stride + r1; if (p < Lin) b1 = inb[p]; }
    v2f av = {a0, a1};
    v2f bv = {b0, b1};
    acc = __builtin_amdgcn_wmma_f32_16x16x4_f32(false, av, false, bv,
                                                (short)0, acc, false, false);
  }
  if (colOK) {
    const int ob = col / Tout, ot = col % Tout;
    const int mbase = half * 8;
#pragma unroll
    for (int v = 0; v < 8; ++v) {
      const int oc = (ty << 4) + mbase + v;
      out[((size_t)ob * Cout + oc) * Tout + ot] = acc[v] + bias[oc];
    }
  }
}

// ---------------------------------------------------------------------------
// GroupNorm(1, C) over (C, L) per sample + optional affine + ReLU (in-place)
// ---------------------------------------------------------------------------
__global__ __launch_bounds__(1024)
void gn_relu(const float* __restrict__ x, float* __restrict__ y,
             const float* __restrict__ g, const float* __restrict__ bt,
             int C, int L)
{
  const int b = blockIdx.x;
  const size_t n = (size_t)C * L;
  const float* xb = x + (size_t)b * n;
  float* yb = y + (size_t)b * n;

  float s = 0.f, s2 = 0.f;
  for (size_t i = threadIdx.x; i < n; i += blockDim.x) {
    float v = xb[i]; s += v; s2 += v * v;
  }
  __shared__ float rs[1024];
  __shared__ float rq[1024];
  rs[threadIdx.x] = s; rq[threadIdx.x] = s2;
  __syncthreads();
  for (int off = blockDim.x >> 1; off > 0; off >>= 1) {
    if ((int)threadIdx.x < off) {
      rs[threadIdx.x] += rs[threadIdx.x + off];
      rq[threadIdx.x] += rq[threadIdx.x + off];
    }
    __syncthreads();
  }
  const float inv_n = 1.0f / (float)n;
  const float mu   = rs[0] * inv_n;
  const float var  = fmaxf(rq[0] * inv_n - mu * mu, 0.f);
  const float rinv = rsqrtf(var + 1e-5f);

  for (size_t i = threadIdx.x; i < n; i += blockDim.x) {
    int c = (int)(i / (size_t)L);
    float v = (xb[i] - mu) * rinv;
    if (g) v = v * g[c] + bt[c];
    yb[i] = fmaxf(v, 0.f);
  }
}

__global__ void log_compress(const float* __restrict__ h, float* __restrict__ z, int n)
{
  int i = blockIdx.x * blockDim.x + threadIdx.x;
  if (i < n) z[i] = __logf(fabsf(h[i]) + 1.0f);
}

__global__ void gather_neg(const float* __restrict__ z, const int* __restrict__ neg,
                           float* __restrict__ zn, int B, int C, int T)
{
  int i = blockIdx.x * blockDim.x + threadIdx.x;
  int total = B * C * T;
  if (i >= total) return;
  int t = i % T;
  int c = (i / T) % C;
  int b = i / (T * C);
  int nv = neg[b * T + t];
  nv += (nv >= t) ? 1 : 0;
  zn[i] = z[((size_t)b * C + c) * T + nv];
}

__global__ void assemble(const float* __restrict__ hk, const float* __restrict__ z,
                         const float* __restrict__ zn, float* __restrict__ out,
                         int B, int C, int T, int P, int KK)
{
  const size_t seg = (size_t)P * B * C * T;
  const size_t total = 3 * seg;
  size_t id = (size_t)blockIdx.x * blockDim.x + threadIdx.x;
  if (id >= total) return;
  const int region = (int)(id / seg);
  size_t q = id % seg;
  const int t = (int)(q % T); q /= T;
  const int c = (int)(q % C); q /= C;
  const int b = (int)(q % B); q /= B;
  const int i = (int)q;
  float v = 0.f;
  if (region == 0) {
    if (i >= 1) v = hk[(((size_t)b * C + c) * T + t) * KK + i];
  } else {
    if (i >= 1 && t >= i + 1) {
      const float* src = (region == 1) ? z : zn;
      v = src[((size_t)b * C + c) * T + t];
    }
  }
  out[id] = v;
}

// ---------------------------------------------------------------------------
extern "C" void kernel_launch(void* const* d_in, const int* in_sizes, int n_in,
                              void* d_out, int out_size, void* d_ws, size_t ws_size,
                              hipStream_t stream)
{
  const float* x       = (const float*)d_in[0];
  const float* ew[5]   = {(const float*)d_in[1], (const float*)d_in[2],
                          (const float*)d_in[3], (const float*)d_in[4],
                          (const float*)d_in[5]};
  const float* encb    = (const float*)d_in[6];
  const float* encg    = (const float*)d_in[7];
  const float* encbeta = (const float*)d_in[8];
  const float* ctxw    = (const float*)d_in[9];
  const float* ctxb    = (const float*)d_in[10];
  const float* ctw     = (const float*)d_in[11];
  const float* ctb     = (const float*)d_in[12];
  const int*   neg     = (const int*)d_in[13];

  const int B = 8, C = 512, KK = 12, P = 11;
  const int L0 = 48000, L1 = 9599, L2 = 2398, L3 = 1198, L4 = 598, T = 298;

  float* ws = (float*)d_ws;
  float* P1 = ws;                 // 8*512*9599 = 39,317,504 (region reused later)
  float* P2 = ws + 39317504;      // 9,822,208
  float* Z  = ws + 49139712;
  float* C1 = Z  + 1220608;
  float* C2 = C1 + 1220608;
  float* ZN = C2 + 1220608;
  float* P3 = ws;                 // reuse region A
  float* P4 = ws + 8000000;
  float* P5 = ws + 12000000;
  float* HK = ws + 16000000;      // 14,647,296

  // ---- encoder layer 1 (direct WMMA) ----
  {
    int tiles = (C / 16) * cdiv_h(B * L1, 16);
    conv1_wmma<<<cdiv_h(tiles, 8), 256, 0, stream>>>(x, ew[0], encb, P1,
                                                     B, L0, C, 5, L1);
    gn_relu<<<B, 1024, 0, stream>>>(P1, P1, encg, encbeta, C, L1);
  }
  // ---- encoder layers 2-5 (LDS-staged WMMA) ----
  {
    int blocks = (C / 64) * cdiv_h(B * L2, 32);
    conv_lds_wmma<512, 8, 0><<<blocks, 256, 0, stream>>>(P1, ew[1], encb + C, P2,
                                                         B, L1, C, 4, L2);
    gn_relu<<<B, 1024, 0, stream>>>(P2, P2, encg + C, encbeta + C, C, L2);
  }
  {
    int blocks = (C / 64) * cdiv_h(B * L3, 32);
    conv_lds_wmma<512, 4, 0><<<blocks, 256, 0, stream>>>(P2, ew[2], encb + 2 * C, P3,
                                                         B, L2, C, 2, L3);
    gn_relu<<<B, 1024, 0, stream>>>(P3, P3, encg + 2 * C, encbeta + 2 * C, C, L3);
  }
  {
    int blocks = (C / 64) * cdiv_h(B * L4, 32);
    conv_lds_wmma<512, 4, 0><<<blocks, 256, 0, stream>>>(P3, ew[3], encb + 3 * C, P4,
                                                         B, L3, C, 2, L4);
    gn_relu<<<B, 1024, 0, stream>>>(P4, P4, encg + 3 * C, encbeta + 3 * C, C, L4);
  }
  {
    int blocks = (C / 64) * cdiv_h(B * T, 32);
    conv_lds_wmma<512, 4, 0><<<blocks, 256, 0, stream>>>(P4, ew[4], encb + 4 * C, P5,
                                                         B, L4, C, 2, T);
    gn_relu<<<B, 1024, 0, stream>>>(P5, P5, encg + 4 * C, encbeta + 4 * C, C, T);
  }

  // ---- log compression ----
  {
    int n = B * C * T;
    log_compress<<<cdiv_h(n, 256), 256, 0, stream>>>(P5, Z, n);
  }

  // ---- context stack ----
  {
    int blocks = (C / 64) * cdiv_h(B * T, 32);
    const float* src = Z;
    for (int l = 0; l < 10; ++l) {
      float* dst = (l % 2 == 0) ? C1 : C2;
      conv_lds_wmma<512, 3, 1><<<blocks, 256, 0, stream>>>(
          src, ctxw + (size_t)l * C * C * 3, ctxb + l * C, dst, B, T, C, 1, T);
      gn_relu<<<B, 1024, 0, stream>>>(dst, dst, nullptr, nullptr, C, T);
      src = dst;
    }
  }

  // ---- conv-transpose einsum -> hk (b,o,t,k) ----
  {
    int blocks = ((C * KK) / 64) * cdiv_h(B * T, 32);
    convT_lds_wmma<<<blocks, 256, 0, stream>>>(C2, ctw, ctb, HK, B, C, T, KK);
  }

  // ---- negative sampling gather ----
  {
    int n = B * C * T;
    gather_neg<<<cdiv_h(n, 256), 256, 0, stream>>>(Z, neg, ZN, B, C, T);
  }

  // ---- final flat buffers ----
  {
    long long total = 3LL * P * B * C * T;
    int blocks = (int)((total + 255) / 256);
    assemble<<<blocks, 256, 0, stream>>>(HK, Z, ZN, (float*)d_out, B, C, T, P, KK);
  }
}